// ISDA_BCELoss_48275432407103
// MI455X (gfx1250) — compile-verified
//
#include <hip/hip_runtime.h>
#include <math.h>

// ISDA-augmented BCE loss as 4 small fp32 WMMA GEMMs + fused epilogues.
// N=256 samples, C=1024 classes, A=384 features. All fp32 -> V_WMMA_F32_16X16X4_F32.
// Each wave owns a 16x32 output tile (two 16x16 WMMA accumulators sharing the
// A fragment) to double WMMA density per strided A-load.

typedef __attribute__((ext_vector_type(2))) float v2f;
typedef __attribute__((ext_vector_type(8))) float v8f;

#define N_DIM 256
#define C_DIM 1024
#define A_DIM 384

// D = A(16x4) * B(4x16) + C(16x16), all fp32, wave32.
__device__ __forceinline__ v8f wmma4(v2f a, v2f b, v8f c) {
  return __builtin_amdgcn_wmma_f32_16x16x4_f32(
      /*neg_a=*/false, a, /*neg_b=*/false, b,
      /*c_mod=*/(short)0, c, /*reuse_a=*/false, /*reuse_b=*/false);
}

// ---------------------------------------------------------------------------
// Kernel 0: amt[c] = sum_n y[n,c]   (tiny column reduction, coalesced over c)
// ---------------------------------------------------------------------------
__global__ void amt_kernel(const float* __restrict__ y, float* __restrict__ amt) {
  int c = blockIdx.x * blockDim.x + threadIdx.x;
  if (c < C_DIM) {
    float s = 0.f;
    #pragma unroll 4
    for (int n = 0; n < N_DIM; ++n) s += y[n * C_DIM + c];
    amt[c] = s;
  }
}

// ---------------------------------------------------------------------------
// Kernel 1: S1 = y^T f, S2 = y^T f^2, fused new_CV epilogue.
//   ave      = S1/amt_safe
//   var_temp = (S2 - 2*ave*S1 + amt*ave^2)/amt_safe     (valid since y binary)
//   w        = amt/(amt+Amount)  (0 if denom==0)
//   new_CV   = CoV*(1-w) + var_temp*w + w*(1-w)*(Ave-ave)^2
// One wave per 16(C) x 32(A) tile: 4 WMMAs per K-step sharing one A fragment.
// K = N = 256 -> 64 iterations.
// ---------------------------------------------------------------------------
__global__ void newcv_kernel(const float* __restrict__ f, const float* __restrict__ y,
                             const float* __restrict__ CoV, const float* __restrict__ AveP,
                             const float* __restrict__ Amount, const float* __restrict__ amt,
                             float* __restrict__ newCV) {
  const int c0 = blockIdx.x * 16;
  const int a0 = blockIdx.y * 32;
  const int l    = threadIdx.x;   // 0..31
  const int half = l >> 4;        // lane half
  const int lm   = l & 15;

  v8f s1a = {}, s2a = {};   // columns a0 .. a0+15
  v8f s1b = {}, s2b = {};   // columns a0+16 .. a0+31
  for (int n0 = 0; n0 < N_DIM; n0 += 4) {
    // A tile (y^T): A[m][k] = y[n0+k][c0+m]; v0 holds K={0|2}, v1 holds K={1|3}
    const int ka = half * 2;
    v2f a;
    a.x = y[(n0 + ka    ) * C_DIM + c0 + lm];
    a.y = y[(n0 + ka + 1) * C_DIM + c0 + lm];
    // B tiles (f): B[k][j] = f[n0+k][a0+j]; rows striped 2-per-VGPR
    const int rb0 = (n0 + half) * A_DIM + a0 + lm;
    const int rb1 = (n0 + half + 2) * A_DIM + a0 + lm;
    v2f b0, b1;
    b0.x = f[rb0];       b0.y = f[rb1];
    b1.x = f[rb0 + 16];  b1.y = f[rb1 + 16];
    v2f q0, q1;
    q0.x = b0.x * b0.x;  q0.y = b0.y * b0.y;
    q1.x = b1.x * b1.x;  q1.y = b1.y * b1.y;
    // Speculative prefetch of the tile 2 K-steps ahead (OOB tail is dropped).
    __builtin_prefetch(&f[rb0 + 8 * A_DIM], 0, 3);
    s1a = wmma4(a, b0, s1a);
    s2a = wmma4(a, q0, s2a);
    s1b = wmma4(a, b1, s1b);
    s2b = wmma4(a, q1, s2b);
  }

  #pragma unroll
  for (int r = 0; r < 8; ++r) {
    const int c = c0 + r + half * 8;   // C/D layout: vgpr r, lanes 0-15 -> M=r, 16-31 -> M=r+8
    const float amt_c    = amt[c];
    const float amt_safe = (amt_c == 0.f) ? 1.f : amt_c;
    const float denom    = amt_c + Amount[c];
    const float w        = (denom == 0.f) ? 0.f : amt_c / denom;
    #pragma unroll
    for (int t = 0; t < 2; ++t) {
      const int a = a0 + t * 16 + lm;
      const float S1 = t ? s1b[r] : s1a[r];
      const float S2 = t ? s2b[r] : s2a[r];
      const float ave      = S1 / amt_safe;
      const float var_temp = (S2 - 2.f * ave * S1 + amt_c * ave * ave) / amt_safe;
      const float d        = AveP[c * A_DIM + a] - ave;
      newCV[c * A_DIM + a] =
          CoV[c * A_DIM + a] * (1.f - w) + var_temp * w + w * (1.f - w) * d * d;
    }
  }
}

// ---------------------------------------------------------------------------
// Kernel 2: CV_temp[N,A] = y[N,C] @ new_CV[C,A]; K = C = 1024 -> 256 iterations.
// One wave per 16(N) x 32(A) tile: 2 WMMAs sharing the strided A fragment.
// ---------------------------------------------------------------------------
__global__ void cvtemp_kernel(const float* __restrict__ y, const float* __restrict__ newCV,
                              float* __restrict__ CVt) {
  const int n0 = blockIdx.x * 16;
  const int a0 = blockIdx.y * 32;
  const int l    = threadIdx.x;
  const int half = l >> 4;
  const int lm   = l & 15;

  v8f acc0 = {}, acc1 = {};
  for (int c0 = 0; c0 < C_DIM; c0 += 4) {
    v2f a;  // A[m][k] = y[n0+m][c0+k]
    a.x = y[(n0 + lm) * C_DIM + c0 + 2 * half];
    a.y = y[(n0 + lm) * C_DIM + c0 + 2 * half + 1];
    const int rb0 = (c0 + half) * A_DIM + a0 + lm;
    const int rb1 = (c0 + half + 2) * A_DIM + a0 + lm;
    v2f b0, b1;  // B[k][j] = new_CV[c0+k][a0+j]
    b0.x = newCV[rb0];       b0.y = newCV[rb1];
    b1.x = newCV[rb0 + 16];  b1.y = newCV[rb1 + 16];
    __builtin_prefetch(&newCV[rb0 + 8 * A_DIM], 0, 3);
    acc0 = wmma4(a, b0, acc0);
    acc1 = wmma4(a, b1, acc1);
  }

  #pragma unroll
  for (int r = 0; r < 8; ++r) {
    const int n = n0 + r + half * 8;
    CVt[n * A_DIM + a0 + lm]      = acc0[r];
    CVt[n * A_DIM + a0 + 16 + lm] = acc1[r];
  }
}

// ---------------------------------------------------------------------------
// Kernel 3: sigma2[N,C] = CV_temp @ (fc_weight^2)^T, fused with
//   aug    = output + sigma2*ratio*0.5*(1-2*target)
//   lterm  = max(aug,0) - aug*target + log1p(exp(-|aug|))
// and an in-wave fixed-order reduction -> one partial per 16x32 tile.
// K = A = 384 -> 96 iterations. One wave per 16(N) x 32(C) tile.
// ---------------------------------------------------------------------------
__global__ void loss_kernel(const float* __restrict__ CVt, const float* __restrict__ W,
                            const float* __restrict__ target, const float* __restrict__ output,
                            const float* __restrict__ ratio_p, float* __restrict__ partials) {
  const int n0 = blockIdx.x * 16;
  const int c0 = blockIdx.y * 32;
  const int l    = threadIdx.x;
  const int half = l >> 4;
  const int lm   = l & 15;

  v8f acc0 = {}, acc1 = {};
  for (int k0 = 0; k0 < A_DIM; k0 += 4) {
    v2f a;  // A[m][k] = CV_temp[n0+m][k0+k]
    a.x = CVt[(n0 + lm) * A_DIM + k0 + 2 * half];
    a.y = CVt[(n0 + lm) * A_DIM + k0 + 2 * half + 1];
    // B[k][j] = fc_weight[c0+j][k0+k]^2  (square on the fly)
    const float w00 = W[(c0 + lm) * A_DIM + k0 + half];
    const float w01 = W[(c0 + lm) * A_DIM + k0 + half + 2];
    const float w10 = W[(c0 + 16 + lm) * A_DIM + k0 + half];
    const float w11 = W[(c0 + 16 + lm) * A_DIM + k0 + half + 2];
    v2f b0, b1;
    b0.x = w00 * w00;  b0.y = w01 * w01;
    b1.x = w10 * w10;  b1.y = w11 * w11;
    acc0 = wmma4(a, b0, acc0);
    acc1 = wmma4(a, b1, acc1);
  }

  const float scale = 0.5f * ratio_p[0];
  float lsum = 0.f;
  #pragma unroll
  for (int r = 0; r < 8; ++r) {
    const int n = n0 + r + half * 8;
    #pragma unroll
    for (int t = 0; t < 2; ++t) {
      const int c = c0 + t * 16 + lm;
      const float s2  = (t ? acc1[r] : acc0[r]) * scale;
      const float tg  = target[n * C_DIM + c];
      const float aug = output[n * C_DIM + c] + s2 * (1.f - 2.f * tg);
      lsum += fmaxf(aug, 0.f) - aug * tg + log1pf(expf(-fabsf(aug)));
    }
  }

  __shared__ float red[32];
  red[l] = lsum;
  __syncthreads();
  if (l == 0) {
    float s = 0.f;
    #pragma unroll
    for (int i = 0; i < 32; ++i) s += red[i];  // fixed order -> deterministic
    partials[blockIdx.y * gridDim.x + blockIdx.x] = s;
  }
}

// ---------------------------------------------------------------------------
// Kernel 4: deterministic final reduction of the partials -> mean loss.
// ---------------------------------------------------------------------------
__global__ void reduce_kernel(const float* __restrict__ partials, int np,
                              float* __restrict__ out) {
  const int l = threadIdx.x;  // 32 threads
  float s = 0.f;
  for (int i = l; i < np; i += 32) s += partials[i];  // fixed strided order
  __shared__ float red[32];
  red[l] = s;
  __syncthreads();
  if (l == 0) {
    float tot = 0.f;
    #pragma unroll
    for (int i = 0; i < 32; ++i) tot += red[i];
    out[0] = tot / (float)(N_DIM * C_DIM);
  }
}

extern "C" void kernel_launch(void* const* d_in, const int* in_sizes, int n_in,
                              void* d_out, int out_size, void* d_ws, size_t ws_size,
                              hipStream_t stream) {
  const float* features = (const float*)d_in[0];  // [N, A]
  const float* target   = (const float*)d_in[1];  // [N, C] binary
  const float* output   = (const float*)d_in[2];  // [N, C]
  const float* fc_w     = (const float*)d_in[3];  // [C, A]
  const float* CoV      = (const float*)d_in[4];  // [C, A]
  const float* AveP     = (const float*)d_in[5];  // [C, A]
  const float* Amount   = (const float*)d_in[6];  // [C]
  const float* ratio    = (const float*)d_in[7];  // scalar

  float* out = (float*)d_out;
  float* ws  = (float*)d_ws;
  // Workspace layout (floats): amt[1024] | new_CV[C*A] | CV_temp[N*A] | partials[512]
  float* amt      = ws;
  float* newCV    = amt + C_DIM;
  float* CVt      = newCV + (size_t)C_DIM * A_DIM;
  float* partials = CVt + (size_t)N_DIM * A_DIM;
  // total = 1024 + 393216 + 98304 + 512 floats ~= 1.93 MB

  amt_kernel<<<dim3((C_DIM + 255) / 256), 256, 0, stream>>>(target, amt);
  newcv_kernel<<<dim3(C_DIM / 16, A_DIM / 32), 32, 0, stream>>>(
      features, target, CoV, AveP, Amount, amt, newCV);
  cvtemp_kernel<<<dim3(N_DIM / 16, A_DIM / 32), 32, 0, stream>>>(target, newCV, CVt);
  loss_kernel<<<dim3(N_DIM / 16, C_DIM / 32), 32, 0, stream>>>(
      CVt, fc_w, target, output, ratio, partials);
  reduce_kernel<<<1, 32, 0, stream>>>(
      partials, (N_DIM / 16) * (C_DIM / 32), out);
}